// GATLayer_2576980378150
// MI455X (gfx1250) — compile-verified
//
#include <hip/hip_runtime.h>

#define N_NODES  50000
#define N_HEDGES 10000
#define N_INC    500000

typedef __attribute__((ext_vector_type(2))) float v2f;
typedef __attribute__((ext_vector_type(8))) float v8f;

// ---------------------------------------------------------------------------
// zero fill
// ---------------------------------------------------------------------------
__global__ void k_zero(float* p, int n) {
    int i = blockIdx.x * blockDim.x + threadIdx.x;
    if (i < n) p[i] = 0.0f;
}

// ---------------------------------------------------------------------------
// collapse:  wc[k,h] = sum_d W[k, h*64+d] * attn[h*64+d]   (W rows are 128 wide)
// ---------------------------------------------------------------------------
__global__ void k_collapse(const float* __restrict__ W, const float* __restrict__ attn,
                           float* __restrict__ wc, int K) {
    int k = blockIdx.x * blockDim.x + threadIdx.x;
    if (k >= K) return;
    for (int h = 0; h < 2; ++h) {
        float s = 0.f;
        const float* wr = W + (size_t)k * 128 + h * 64;
        const float* ar = attn + h * 64;
        for (int d = 0; d < 64; ++d) s += wr[d] * ar[d];
        wc[k * 2 + h] = s;
    }
}

// ---------------------------------------------------------------------------
// WMMA GEMM:  C[M,128] = A[M,K] * B[K,128]   (f32, V_WMMA_F32_16X16X4_F32)
// grid.x = M/16 (exact), block = 256 = 8 waves, wave w -> column tile w
// ---------------------------------------------------------------------------
template <int K>
__global__ void k_wmma_gemm(const float* __restrict__ A, const float* __restrict__ B,
                            float* __restrict__ C) {
    const int lane = threadIdx.x & 31;
    const int wv   = threadIdx.x >> 5;          // 0..7 -> N tile
    const int col  = wv * 16 + (lane & 15);     // output column
    const int arow = blockIdx.x * 16 + (lane & 15);
    const int kb   = (lane >> 4) * 2;           // 0 or 2 (K sub-offset per half-wave)

    v8f acc = {};
#pragma unroll
    for (int k = 0; k < K; k += 4) {
        float2 a2 = *(const float2*)(A + (size_t)arow * K + k + kb);
        float  b0 = B[(size_t)(k + kb) * 128 + col];
        float  b1 = B[(size_t)(k + kb + 1) * 128 + col];
        v2f av = {a2.x, a2.y};
        v2f bv = {b0, b1};
        acc = __builtin_amdgcn_wmma_f32_16x16x4_f32(false, av, false, bv,
                                                    (short)0, acc, false, false);
    }
    // D layout: VGPR i holds M = i + 8*(lane>>4), N = lane&15 (within tile)
    const int mbase = blockIdx.x * 16 + 8 * (lane >> 4);
#pragma unroll
    for (int i = 0; i < 8; ++i)
        C[(size_t)(mbase + i) * 128 + col] = acc[i];
}

// ---------------------------------------------------------------------------
// matvec2: y[r,h] = sum_k X[r,k] * wc[k,h]   (one wave32 per row)
// ---------------------------------------------------------------------------
template <int K>
__global__ void k_matvec2(const float* __restrict__ X, const float* __restrict__ wc,
                          float* __restrict__ y, int R) {
    int row  = blockIdx.x * (blockDim.x >> 5) + (threadIdx.x >> 5);
    int lane = threadIdx.x & 31;
    if (row >= R) return;
    const int per = K / 32;
    float s0 = 0.f, s1 = 0.f;
#pragma unroll
    for (int j = 0; j < per; ++j) {
        int k = lane * per + j;
        float x = X[(size_t)row * K + k];
        s0 += x * wc[k * 2 + 0];
        s1 += x * wc[k * 2 + 1];
    }
#pragma unroll
    for (int off = 16; off > 0; off >>= 1) {
        s0 += __shfl_xor(s0, off, 32);
        s1 += __shfl_xor(s1, off, 32);
    }
    if (lane == 0) { y[row * 2] = s0; y[row * 2 + 1] = s1; }
}

// ---------------------------------------------------------------------------
// edge pass 1: score = relu(L[ia] + Rr[ib]);  segment max via uint atomicMax
// (scores are >=0 so IEEE bits order == unsigned order; mbuf pre-zeroed)
// ---------------------------------------------------------------------------
__global__ void k_edge_max(const int* __restrict__ ia, const int* __restrict__ ib,
                           const float* __restrict__ L, const float* __restrict__ Rr,
                           float* __restrict__ ebuf, float* __restrict__ mbuf, int E) {
    int i = blockIdx.x * blockDim.x + threadIdx.x;
    if (i >= E) return;
    int a = ia[i], b = ib[i];
    float e0 = L[a * 2 + 0] + Rr[b * 2 + 0];
    float e1 = L[a * 2 + 1] + Rr[b * 2 + 1];
    e0 = e0 > 0.f ? e0 : 0.f;
    e1 = e1 > 0.f ? e1 : 0.f;
    ebuf[i * 2 + 0] = e0;
    ebuf[i * 2 + 1] = e1;
    atomicMax((unsigned int*)&mbuf[b * 2 + 0], __float_as_uint(e0));
    atomicMax((unsigned int*)&mbuf[b * 2 + 1], __float_as_uint(e1));
}

// ---------------------------------------------------------------------------
// edge pass 2: ex = exp(e - m[seg]); segment sum of ex (sbuf pre-zeroed)
// ---------------------------------------------------------------------------
__global__ void k_edge_exp(const int* __restrict__ ib, float* __restrict__ ebuf,
                           const float* __restrict__ mbuf, float* __restrict__ sbuf, int E) {
    int i = blockIdx.x * blockDim.x + threadIdx.x;
    if (i >= E) return;
    int b = ib[i];
    float x0 = expf(ebuf[i * 2 + 0] - mbuf[b * 2 + 0]);
    float x1 = expf(ebuf[i * 2 + 1] - mbuf[b * 2 + 1]);
    ebuf[i * 2 + 0] = x0;
    ebuf[i * 2 + 1] = x1;
    atomicAdd(&sbuf[b * 2 + 0], x0);
    atomicAdd(&sbuf[b * 2 + 1], x1);
}

// ---------------------------------------------------------------------------
// edge pass 3: agg[seg, h, d] += F[idxF, h, d] * ex/s    (one wave32 per edge,
// each lane owns a float4 chunk of the 128-wide feature row)
// ---------------------------------------------------------------------------
__global__ void k_edge_scatter(const int* __restrict__ idxF, const int* __restrict__ seg,
                               const float* __restrict__ F, const float* __restrict__ exb,
                               const float* __restrict__ sbuf, float* __restrict__ agg, int E) {
    int i = blockIdx.x * 8 + (threadIdx.x >> 5);
    if (i >= E) return;
    int j = threadIdx.x & 31;      // float4 chunk id, 32*4 = 128 features
    int h = j >> 4;                // head: chunks 0-15 -> h0, 16-31 -> h1
    int fi = idxF[i], sg = seg[i];
    float a = exb[i * 2 + h] / sbuf[sg * 2 + h];
    float4 v = *(const float4*)(F + (size_t)fi * 128 + j * 4);
    float* dst = agg + (size_t)sg * 128 + j * 4;
    atomicAdd(dst + 0, v.x * a);
    atomicAdd(dst + 1, v.y * a);
    atomicAdd(dst + 2, v.z * a);
    atomicAdd(dst + 3, v.w * a);
}

// ---------------------------------------------------------------------------
// finalize: out[r,d] = 0.5*((agg[r,0,d]+bias[d]) + (agg[r,1,d]+bias[64+d]))
// ---------------------------------------------------------------------------
__global__ void k_finalize(const float* __restrict__ agg, const float* __restrict__ bias,
                           float* __restrict__ out, int R) {
    int t = blockIdx.x * blockDim.x + threadIdx.x;
    if (t >= R * 64) return;
    int r = t >> 6, d = t & 63;
    float s = (agg[(size_t)r * 128 + d] + bias[d]) +
              (agg[(size_t)r * 128 + 64 + d] + bias[64 + d]);
    out[t] = 0.5f * s;
}

// ---------------------------------------------------------------------------
extern "C" void kernel_launch(void* const* d_in, const int* in_sizes, int n_in,
                              void* d_out, int out_size, void* d_ws, size_t ws_size,
                              hipStream_t stream) {
    (void)in_sizes; (void)n_in; (void)out_size; (void)ws_size;

    const float* vfeat   = (const float*)d_in[0];
    const float* efeat   = (const float*)d_in[1];
    const int*   inc_n   = (const int*)d_in[2];
    const int*   inc_h   = (const int*)d_in[3];
    const float* W_ke    = (const float*)d_in[4];
    const float* W_qe    = (const float*)d_in[5];
    const float* attn_ke = (const float*)d_in[6];
    const float* attn_qe = (const float*)d_in[7];
    const float* bias_e  = (const float*)d_in[8];
    const float* W_kv    = (const float*)d_in[9];
    const float* W_qv    = (const float*)d_in[10];
    const float* attn_kv = (const float*)d_in[11];
    const float* attn_qv = (const float*)d_in[12];
    const float* bias_v  = (const float*)d_in[13];

    float* out   = (float*)d_out;
    float* vout  = out;                       // [50000,64]
    float* eout  = out + (size_t)N_NODES * 64; // [10000,64]

    // workspace layout (floats); reg1/reg2 aliased across phases
    float* ws   = (float*)d_ws;
    float* reg1 = ws;                   // 6,400,000 : vke (ph1) / aggV (ph2)
    float* reg2 = reg1 + 6400000;       // 1,280,000 : aggE (ph1) / ekv (ph2)
    float* ebuf = reg2 + 1280000;       // 1,000,000 : e then ex, [E,2]
    float* el   = ebuf + 1000000;       // 100,000
    float* er   = el + 100000;          //  20,000
    float* el2  = er + 20000;           //  20,000
    float* er2  = el2 + 20000;          // 100,000
    float* mbuf = er2 + 100000;         // 100,000
    float* sbuf = mbuf + 100000;        // 100,000
    float* wkeC = sbuf + 100000;        // 256
    float* wqeC = wkeC + 256;           // 256
    float* wkvC = wqeC + 256;           // 256
    float* wqvC = wkvC + 256;           // 256

    const int E = N_INC;
    dim3 b256(256);

    // ================= Phase 1: nodes -> hyperedges =================
    k_zero<<<dim3((200000 + 255) / 256), b256, 0, stream>>>(mbuf, 200000);        // mbuf+sbuf
    k_zero<<<dim3((1280000 + 255) / 256), b256, 0, stream>>>(reg2, 1280000);      // aggE

    k_collapse<<<dim3(1), dim3(128), 0, stream>>>(W_ke, attn_ke, wkeC, 128);
    k_collapse<<<dim3(1), dim3(128), 0, stream>>>(W_qe, attn_qe, wqeC, 128);

    // vke = vfeat @ W_ke   (50000 x 128) -- WMMA
    k_wmma_gemm<128><<<dim3(N_NODES / 16), b256, 0, stream>>>(vfeat, W_ke, reg1);

    k_matvec2<128><<<dim3(N_NODES / 8), b256, 0, stream>>>(vfeat, wkeC, el, N_NODES);
    k_matvec2<128><<<dim3(N_HEDGES / 8), b256, 0, stream>>>(efeat, wqeC, er, N_HEDGES);

    k_edge_max<<<dim3((E + 255) / 256), b256, 0, stream>>>(inc_n, inc_h, el, er, ebuf, mbuf, E);
    k_edge_exp<<<dim3((E + 255) / 256), b256, 0, stream>>>(inc_h, ebuf, mbuf, sbuf, E);
    k_edge_scatter<<<dim3((E + 7) / 8), b256, 0, stream>>>(inc_n, inc_h, reg1, ebuf, sbuf, reg2, E);

    k_finalize<<<dim3((N_HEDGES * 64 + 255) / 256), b256, 0, stream>>>(reg2, bias_e, eout, N_HEDGES);

    // ================= Phase 2: hyperedges -> nodes =================
    k_zero<<<dim3((200000 + 255) / 256), b256, 0, stream>>>(mbuf, 200000);        // mbuf+sbuf
    k_zero<<<dim3((6400000 + 255) / 256), b256, 0, stream>>>(reg1, 6400000);      // aggV

    k_collapse<<<dim3(1), dim3(128), 0, stream>>>(W_kv, attn_kv, wkvC, 64);
    k_collapse<<<dim3(1), dim3(128), 0, stream>>>(W_qv, attn_qv, wqvC, 128);

    // ekv = eout @ W_kv   (10000 x 128, K=64) -- WMMA
    k_wmma_gemm<64><<<dim3(N_HEDGES / 16), b256, 0, stream>>>(eout, W_kv, reg2);

    k_matvec2<64><<<dim3(N_HEDGES / 8), b256, 0, stream>>>(eout, wkvC, el2, N_HEDGES);
    k_matvec2<128><<<dim3(N_NODES / 8), b256, 0, stream>>>(vfeat, wqvC, er2, N_NODES);

    k_edge_max<<<dim3((E + 255) / 256), b256, 0, stream>>>(inc_h, inc_n, el2, er2, ebuf, mbuf, E);
    k_edge_exp<<<dim3((E + 255) / 256), b256, 0, stream>>>(inc_n, ebuf, mbuf, sbuf, E);
    k_edge_scatter<<<dim3((E + 7) / 8), b256, 0, stream>>>(inc_h, inc_n, reg2, ebuf, sbuf, reg1, E);

    k_finalize<<<dim3((N_NODES * 64 + 255) / 256), b256, 0, stream>>>(reg1, bias_v, vout, N_NODES);
}